// HyperGAT_88055419503323
// MI455X (gfx1250) — compile-verified
//
#include <hip/hip_runtime.h>
#include <hip/hip_bf16.h>
#include <math.h>

#define N_NODES 100000
#define N_EDGES 100000
#define NNZ     3200000
#define D_IN    512
#define D_HID   16
#define N_CLS   7
#define NEG_SLOPE 0.2f

typedef __attribute__((ext_vector_type(2))) float v2f;
typedef __attribute__((ext_vector_type(8))) float v8f;

// ---------- helpers ----------
__device__ __forceinline__ unsigned f2ord(float f) {
    unsigned u = __float_as_uint(f);
    return (u & 0x80000000u) ? ~u : (u | 0x80000000u);
}
__device__ __forceinline__ float ord2f(unsigned u) {
    return (u & 0x80000000u) ? __uint_as_float(u & 0x7fffffffu)
                             : __uint_as_float(~u);
}
__device__ __forceinline__ float lrelu(float x) { return x > 0.f ? x : NEG_SLOPE * x; }
__device__ __forceinline__ float elu(float x)   { return x > 0.f ? x : __expf(x) - 1.f; }

__global__ void zero_f32(float* p, int n) {
    int i = blockIdx.x * blockDim.x + threadIdx.x;
    if (i < n) p[i] = 0.f;
}

// ---------- GEMM 1: h = H @ W1   [N,512]x[512,16] -> [N,16]  (f32 WMMA 16x16x4) ----------
__global__ void __launch_bounds__(256)
gemm1_wmma(const float* __restrict__ H, const float* __restrict__ W,
           float* __restrict__ out) {
    __shared__ float sW[D_IN * D_HID]; // 32 KB
    for (int i = threadIdx.x; i < D_IN * D_HID; i += blockDim.x) sW[i] = W[i];
    __syncthreads();

    const int lane = threadIdx.x & 31;
    const int wave = threadIdx.x >> 5;
    const int tile = blockIdx.x * (blockDim.x >> 5) + wave; // 16-row tile
    if (tile >= N_NODES / 16) return;

    const int mrow = tile * 16 + (lane & 15);
    const int koff = (lane >> 4) * 2;   // lanes 16-31 hold K+2,K+3
    const int ncol = lane & 15;

    v8f acc = {};
    const float* arow = H + (size_t)mrow * D_IN + koff;
    #pragma unroll 8
    for (int k0 = 0; k0 < D_IN; k0 += 4) {
        v2f a = *(const v2f*)(arow + k0);
        v2f b;
        b.x = sW[(k0 + koff + 0) * D_HID + ncol];
        b.y = sW[(k0 + koff + 1) * D_HID + ncol];
        acc = __builtin_amdgcn_wmma_f32_16x16x4_f32(false, a, false, b,
                                                    (short)0, acc, false, false);
    }
    // D layout: VGPR r -> M=r (lanes 0-15) / M=r+8 (lanes 16-31); N = lane&15
    const int mbase = tile * 16 + ((lane >> 4) ? 8 : 0);
    #pragma unroll
    for (int r = 0; r < 8; ++r)
        out[(size_t)(mbase + r) * D_HID + ncol] = acc[r];
}

// ---------- GEMM 2: h2 = elu(H1) @ W2   [N,16]x[16,7] -> [N,7]  (WMMA, N padded to 16) ----------
__global__ void __launch_bounds__(256)
gemm2_wmma(const float* __restrict__ H1, const float* __restrict__ W2,
           float* __restrict__ out) {
    __shared__ float sW[16 * 16];
    if (threadIdx.x < 256) {
        int r = threadIdx.x >> 4, c = threadIdx.x & 15;
        sW[threadIdx.x] = (c < N_CLS) ? W2[r * N_CLS + c] : 0.f;
    }
    __syncthreads();

    const int lane = threadIdx.x & 31;
    const int wave = threadIdx.x >> 5;
    const int tile = blockIdx.x * (blockDim.x >> 5) + wave;
    if (tile >= N_NODES / 16) return;

    const int mrow = tile * 16 + (lane & 15);
    const int koff = (lane >> 4) * 2;
    const int ncol = lane & 15;

    v8f acc = {};
    const float* arow = H1 + (size_t)mrow * D_HID + koff;
    #pragma unroll
    for (int k0 = 0; k0 < D_HID; k0 += 4) {
        v2f a;
        a.x = elu(arow[k0 + 0]);
        a.y = elu(arow[k0 + 1]);
        v2f b;
        b.x = sW[(k0 + koff + 0) * 16 + ncol];
        b.y = sW[(k0 + koff + 1) * 16 + ncol];
        acc = __builtin_amdgcn_wmma_f32_16x16x4_f32(false, a, false, b,
                                                    (short)0, acc, false, false);
    }
    const int mbase = tile * 16 + ((lane >> 4) ? 8 : 0);
    if (ncol < N_CLS) {
        #pragma unroll
        for (int r = 0; r < 8; ++r)
            out[(size_t)(mbase + r) * N_CLS + ncol] = acc[r];
    }
}

// ---------- attention pipeline (templated on feature width D) ----------
// per-node scalars: t1[v] = h[v]·a1 ; t2h[v] = h[v]·a2[0:D]
template <int D>
__global__ void node_scalars(const float* __restrict__ h, const float* __restrict__ a1,
                             const float* __restrict__ a2, float* __restrict__ t1,
                             float* __restrict__ t2h) {
    int v = blockIdx.x * blockDim.x + threadIdx.x;
    if (v >= N_NODES) return;
    float s1 = 0.f, s2 = 0.f;
    #pragma unroll
    for (int d = 0; d < D; ++d) {
        float x = h[(size_t)v * D + d];
        s1 += x * a1[d];
        s2 += x * a2[d];
    }
    t1[v] = s1;
    t2h[v] = s2;
}

__global__ void edge_max(const int* __restrict__ ni, const int* __restrict__ ei,
                         const float* __restrict__ t1, unsigned* __restrict__ m1) {
    int j = blockIdx.x * blockDim.x + threadIdx.x;
    if (j >= NNZ) return;
    float s = lrelu(t1[ni[j]]);
    atomicMax(&m1[ei[j]], f2ord(s));
}

__global__ void edge_sum(const int* __restrict__ ni, const int* __restrict__ ei,
                         const float* __restrict__ t1, const unsigned* __restrict__ m1,
                         float* __restrict__ z1) {
    int j = blockIdx.x * blockDim.x + threadIdx.x;
    if (j >= NNZ) return;
    int e = ei[j];
    float s = lrelu(t1[ni[j]]);
    atomicAdd(&z1[e], __expf(s - ord2f(m1[e])));
}

template <int D>
__global__ void edge_agg(const int* __restrict__ ni, const int* __restrict__ ei,
                         const float* __restrict__ t1, const unsigned* __restrict__ m1,
                         const float* __restrict__ z1, const float* __restrict__ h,
                         float* __restrict__ eacc) {
    int j = blockIdx.x * blockDim.x + threadIdx.x;
    if (j >= NNZ) return;
    int v = ni[j], e = ei[j];
    float s = lrelu(t1[v]);
    float alpha = __expf(s - ord2f(m1[e])) / (z1[e] + 1e-9f);
    #pragma unroll
    for (int d = 0; d < D; ++d)
        atomicAdd(&eacc[(size_t)e * D + d], alpha * h[(size_t)v * D + d]);
}

// per-edge scalar: t2e[e] = eacc[e]·a2[D:2D]
template <int D>
__global__ void edge_scalar(const float* __restrict__ eacc, const float* __restrict__ a2,
                            float* __restrict__ t2e) {
    int e = blockIdx.x * blockDim.x + threadIdx.x;
    if (e >= N_EDGES) return;
    float s = 0.f;
    #pragma unroll
    for (int d = 0; d < D; ++d) s += eacc[(size_t)e * D + d] * a2[D + d];
    t2e[e] = s;
}

__global__ void node_max(const int* __restrict__ ni, const int* __restrict__ ei,
                         const float* __restrict__ t2h, const float* __restrict__ t2e,
                         unsigned* __restrict__ m2) {
    int j = blockIdx.x * blockDim.x + threadIdx.x;
    if (j >= NNZ) return;
    int v = ni[j];
    float s = lrelu(t2h[v] + t2e[ei[j]]);
    atomicMax(&m2[v], f2ord(s));
}

__global__ void node_sum(const int* __restrict__ ni, const int* __restrict__ ei,
                         const float* __restrict__ t2h, const float* __restrict__ t2e,
                         const unsigned* __restrict__ m2, float* __restrict__ z2) {
    int j = blockIdx.x * blockDim.x + threadIdx.x;
    if (j >= NNZ) return;
    int v = ni[j];
    float s = lrelu(t2h[v] + t2e[ei[j]]);
    atomicAdd(&z2[v], __expf(s - ord2f(m2[v])));
}

template <int D>
__global__ void node_agg(const int* __restrict__ ni, const int* __restrict__ ei,
                         const float* __restrict__ t2h, const float* __restrict__ t2e,
                         const unsigned* __restrict__ m2, const float* __restrict__ z2,
                         const float* __restrict__ eacc, float* __restrict__ out) {
    int j = blockIdx.x * blockDim.x + threadIdx.x;
    if (j >= NNZ) return;
    int v = ni[j], e = ei[j];
    float s = lrelu(t2h[v] + t2e[e]);
    float beta = __expf(s - ord2f(m2[v])) / (z2[v] + 1e-9f);
    #pragma unroll
    for (int d = 0; d < D; ++d)
        atomicAdd(&out[(size_t)v * D + d], beta * eacc[(size_t)e * D + d]);
}

__global__ void log_softmax7(const float* __restrict__ H2, float* __restrict__ logp) {
    int v = blockIdx.x * blockDim.x + threadIdx.x;
    if (v >= N_NODES) return;
    float x[N_CLS];
    float m = -INFINITY;
    #pragma unroll
    for (int c = 0; c < N_CLS; ++c) { x[c] = H2[(size_t)v * N_CLS + c]; m = fmaxf(m, x[c]); }
    float s = 0.f;
    #pragma unroll
    for (int c = 0; c < N_CLS; ++c) s += __expf(x[c] - m);
    float l = m + __logf(s);
    #pragma unroll
    for (int c = 0; c < N_CLS; ++c) logp[(size_t)v * N_CLS + c] = x[c] - l;
}

// ---------- host side ----------
static inline int cdiv(int a, int b) { return (a + b - 1) / b; }

extern "C" void kernel_launch(void* const* d_in, const int* in_sizes, int n_in,
                              void* d_out, int out_size, void* d_ws, size_t ws_size,
                              hipStream_t stream) {
    const float* H    = (const float*)d_in[0];
    const float* W1   = (const float*)d_in[1];
    const float* a1_1 = (const float*)d_in[2];
    const float* a2_1 = (const float*)d_in[3];
    const float* W2   = (const float*)d_in[4];
    const float* a1_2 = (const float*)d_in[5];
    const float* a2_2 = (const float*)d_in[6];
    const int*   ni   = (const int*)d_in[7];
    const int*   ei   = (const int*)d_in[8];

    float* logp  = (float*)d_out;                       // [N,7]
    float* H1out = logp + (size_t)N_NODES * N_CLS;      // [N,16]
    float* H2out = H1out + (size_t)N_NODES * D_HID;     // [N,7]

    // workspace layout (floats)
    float*    ws   = (float*)d_ws;
    float*    hbuf = ws;                           // N*16 (layer1 GEMM out; reused N*7 layer2)
    float*    eacc = hbuf + (size_t)N_NODES * 16;  // E*16 (reused E*7 layer2)
    float*    t1   = eacc + (size_t)N_EDGES * 16;  // N
    float*    t2h  = t1 + N_NODES;                 // N
    float*    t2e  = t2h + N_NODES;                // E
    unsigned* m1   = (unsigned*)(t2e + N_EDGES);   // E
    float*    z1   = (float*)(m1 + N_EDGES);       // E
    unsigned* m2   = (unsigned*)(z1 + N_EDGES);    // N
    float*    z2   = (float*)(m2 + N_NODES);       // N

    const int B = 256;
    const int gNode = cdiv(N_NODES, B);
    const int gEdge = cdiv(N_EDGES, B);
    const int gNnz  = cdiv(NNZ, B);
    const int gTile = cdiv(N_NODES / 16, B / 32); // 8 waves per block

    // ----- layer 1 -----
    gemm1_wmma<<<gTile, B, 0, stream>>>(H, W1, hbuf);

    zero_f32<<<gEdge, B, 0, stream>>>((float*)m1, N_EDGES);
    zero_f32<<<gEdge, B, 0, stream>>>(z1, N_EDGES);
    zero_f32<<<cdiv(N_EDGES * D_HID, B), B, 0, stream>>>(eacc, N_EDGES * D_HID);
    zero_f32<<<gNode, B, 0, stream>>>((float*)m2, N_NODES);
    zero_f32<<<gNode, B, 0, stream>>>(z2, N_NODES);
    zero_f32<<<cdiv(N_NODES * D_HID, B), B, 0, stream>>>(H1out, N_NODES * D_HID);

    node_scalars<D_HID><<<gNode, B, 0, stream>>>(hbuf, a1_1, a2_1, t1, t2h);
    edge_max<<<gNnz, B, 0, stream>>>(ni, ei, t1, m1);
    edge_sum<<<gNnz, B, 0, stream>>>(ni, ei, t1, m1, z1);
    edge_agg<D_HID><<<gNnz, B, 0, stream>>>(ni, ei, t1, m1, z1, hbuf, eacc);
    edge_scalar<D_HID><<<gEdge, B, 0, stream>>>(eacc, a2_1, t2e);
    node_max<<<gNnz, B, 0, stream>>>(ni, ei, t2h, t2e, m2);
    node_sum<<<gNnz, B, 0, stream>>>(ni, ei, t2h, t2e, m2, z2);
    node_agg<D_HID><<<gNnz, B, 0, stream>>>(ni, ei, t2h, t2e, m2, z2, eacc, H1out);

    // ----- layer 2 -----
    gemm2_wmma<<<gTile, B, 0, stream>>>(H1out, W2, hbuf); // hbuf now [N,7]

    zero_f32<<<gEdge, B, 0, stream>>>((float*)m1, N_EDGES);
    zero_f32<<<gEdge, B, 0, stream>>>(z1, N_EDGES);
    zero_f32<<<cdiv(N_EDGES * N_CLS, B), B, 0, stream>>>(eacc, N_EDGES * N_CLS);
    zero_f32<<<gNode, B, 0, stream>>>((float*)m2, N_NODES);
    zero_f32<<<gNode, B, 0, stream>>>(z2, N_NODES);
    zero_f32<<<cdiv(N_NODES * N_CLS, B), B, 0, stream>>>(H2out, N_NODES * N_CLS);

    node_scalars<N_CLS><<<gNode, B, 0, stream>>>(hbuf, a1_2, a2_2, t1, t2h);
    edge_max<<<gNnz, B, 0, stream>>>(ni, ei, t1, m1);
    edge_sum<<<gNnz, B, 0, stream>>>(ni, ei, t1, m1, z1);
    edge_agg<N_CLS><<<gNnz, B, 0, stream>>>(ni, ei, t1, m1, z1, hbuf, eacc);
    edge_scalar<N_CLS><<<gEdge, B, 0, stream>>>(eacc, a2_2, t2e);
    node_max<<<gNnz, B, 0, stream>>>(ni, ei, t2h, t2e, m2);
    node_sum<<<gNnz, B, 0, stream>>>(ni, ei, t2h, t2e, m2, z2);
    node_agg<N_CLS><<<gNnz, B, 0, stream>>>(ni, ei, t2h, t2e, m2, z2, eacc, H2out);

    // ----- output head -----
    log_softmax7<<<gNode, B, 0, stream>>>(H2out, logp);
}